// Pooling_45354854645954
// MI455X (gfx1250) — compile-verified
//
#include <hip/hip_runtime.h>
#include <stdint.h>

#define B_   8
#define S_   4096
#define D_   768
#define N_   64
#define TPB  256
#define CPT  (D_ / TPB)              // 3 channels per thread
#define TILE 8                       // tokens per LDS tile
#define F4_PER_TILE (TILE * D_ / 4)  // 1536 float4 per tile
#define LPT  (F4_PER_TILE / TPB)     // 6 async b128 loads / thread / tile

static_assert(D_ % TPB == 0, "channel split");
static_assert(LPT == 6, "loads per thread must match wait immediate");

// ---- CDNA5 async global->LDS path (guarded; falls back to sync copy) ----
#if defined(__HIP_DEVICE_COMPILE__) && defined(__gfx1250__) && \
    __has_builtin(__builtin_amdgcn_global_load_async_to_lds_b128)
#define USE_ASYNC 1
#endif

#ifdef USE_ASYNC
#define AS1 __attribute__((address_space(1)))
#define AS3 __attribute__((address_space(3)))
typedef int v4i_t __attribute__((ext_vector_type(4)));
__device__ __forceinline__ void async_cp16(const float4* g, float4* l) {
  // 16B per lane, DMA into LDS, tracked by ASYNCcnt.
  // Builtin signature (probe-confirmed): (v4i AS1*, v4i AS3*, imm offset, imm cpol)
  v4i_t* gv = (v4i_t*)g;   // drop const + reinterpret in generic AS
  v4i_t* lv = (v4i_t*)l;
  __builtin_amdgcn_global_load_async_to_lds_b128((AS1 v4i_t*)gv, (AS3 v4i_t*)lv, 0, 0);
}
#if __has_builtin(__builtin_amdgcn_s_wait_asynccnt)
#define WAIT_ASYNC(nn) __builtin_amdgcn_s_wait_asynccnt(nn)
#else
#define WAIT_ASYNC(nn) asm volatile("s_wait_asynccnt %0" :: "i"(nn) : "memory")
#endif
#else
__device__ __forceinline__ void async_cp16(const float4* g, float4* l) { *l = *g; }
#define WAIT_ASYNC(nn) ((void)0)
#endif

__global__ __launch_bounds__(TPB)
void pool_seg_mean_kernel(const float* __restrict__ wv,        // [B,S,D]
                          const int*   __restrict__ rep_ids,   // [B,N]
                          const float* __restrict__ rep_mask,  // [B,N]
                          const int*   __restrict__ lens,      // [B,N]
                          const float* __restrict__ len_mask,  // [B,N]
                          float*       __restrict__ out)       // [B*2N*D | B*2N]
{
  __shared__ __align__(16) float smem[2][TILE * D_];
  __shared__ int   s_meta[2];
  __shared__ float s_totcnt;

  const int bn  = blockIdx.x;
  const int b   = bn / N_;
  const int n   = bn - b * N_;
  const int tid = threadIdx.x;

  if (tid == 0) {
    const int* lrow = lens + b * N_;
    int st = 0;
    for (int k = 0; k < n; ++k) st += lrow[k];
    s_meta[0] = st;
    s_meta[1] = lrow[n];
    s_totcnt  = 0.0f;
  }
  __syncthreads();
  const int start = s_meta[0];
  const int len   = s_meta[1];   // >= 1 by construction

  const float* base = wv + ((size_t)b * S_ + (size_t)start) * D_;

  float sum[CPT], cnt[CPT];
#pragma unroll
  for (int k = 0; k < CPT; ++k) { sum[k] = 0.0f; cnt[k] = 0.0f; }

  const int nt = (len + TILE - 1) / TILE;

  // Prologue: stream tile 0 into buffer 0 (tail rows clamp source to a valid row)
#pragma unroll
  for (int j = 0; j < LPT; ++j) {
    const int f  = tid + j * TPB;
    const int tt = f / (D_ / 4);
    const int cc = f - tt * (D_ / 4);
    int grow = tt; if (grow > len - 1) grow = len - 1;
    async_cp16((const float4*)(base + (size_t)grow * D_) + cc,
               (float4*)(&smem[0][tt * D_]) + cc);
  }

  for (int t = 0; t < nt; ++t) {
    const int cur = t & 1;
    if (t + 1 < nt) {
      const int nxt   = (t + 1) & 1;
      const int tbase = (t + 1) * TILE;
#pragma unroll
      for (int j = 0; j < LPT; ++j) {
        const int f  = tid + j * TPB;
        const int tt = f / (D_ / 4);
        const int cc = f - tt * (D_ / 4);
        int grow = tbase + tt; if (grow > len - 1) grow = len - 1;
        async_cp16((const float4*)(base + (size_t)grow * D_) + cc,
                   (float4*)(&smem[nxt][tt * D_]) + cc);
      }
      WAIT_ASYNC(6);   // async loads retire in order: <=6 left => tile t landed
    } else {
      WAIT_ASYNC(0);
    }
    __syncthreads();   // everyone's slice of tile t is visible

    const int rem  = len - t * TILE;
    const int rows = rem < TILE ? rem : TILE;
    for (int r = 0; r < rows; ++r) {
#pragma unroll
      for (int k = 0; k < CPT; ++k) {
        const float v = smem[cur][r * D_ + tid + k * TPB];
        sum[k] += v;
        cnt[k] += (v != 0.0f) ? 1.0f : 0.0f;
      }
    }
    __syncthreads();   // done reading buf[cur] before it is refilled at t+2
  }

  // block-total nonzero count (for the all-zero fallback)
  float myc = 0.0f;
#pragma unroll
  for (int k = 0; k < CPT; ++k) myc += cnt[k];
  atomicAdd(&s_totcnt, myc);

  // rep-token gather branch (overlaps with the barrier below)
  const int   id = rep_ids[b * N_ + n];
  const float rm = rep_mask[b * N_ + n];
  const float lm = len_mask[b * N_ + n];
  const float* rrow     = wv + ((size_t)b * S_ + (size_t)id) * D_;
  float*       out_rep  = out + ((size_t)b * 2 * N_ + n) * D_;
  float*       out_mean = out + ((size_t)b * 2 * N_ + N_ + n) * D_;
#pragma unroll
  for (int k = 0; k < CPT; ++k) {
    const int d = tid + k * TPB;
    out_rep[d] = rrow[d] * rm;
  }

  __syncthreads();
  const bool allz = (s_totcnt == 0.0f);
#pragma unroll
  for (int k = 0; k < CPT; ++k) {
    const int d = tid + k * TPB;
    float m = sum[k] / fmaxf(cnt[k], 1.0f);
    if (allz) m = wv[d];          // word_vectors[0,0,:] fallback
    out_mean[d] = m * lm;
  }

  if (tid == 0) {
    float* om = out + (size_t)B_ * 2 * N_ * D_;
    om[b * 2 * N_ + n]      = rm;
    om[b * 2 * N_ + N_ + n] = lm;
  }
}

extern "C" void kernel_launch(void* const* d_in, const int* in_sizes, int n_in,
                              void* d_out, int out_size, void* d_ws, size_t ws_size,
                              hipStream_t stream) {
  (void)in_sizes; (void)n_in; (void)out_size; (void)d_ws; (void)ws_size;
  const float* wv       = (const float*)d_in[0];
  const int*   rep_ids  = (const int*)  d_in[1];
  const float* rep_mask = (const float*)d_in[2];
  const int*   lens     = (const int*)  d_in[3];
  const float* len_mask = (const float*)d_in[4];
  float*       out      = (float*)d_out;

  pool_seg_mean_kernel<<<B_ * N_, TPB, 0, stream>>>(
      wv, rep_ids, rep_mask, lens, len_mask, out);
}